// Model_distance_81707457839662
// MI455X (gfx1250) — compile-verified
//
#include <hip/hip_runtime.h>

typedef __attribute__((ext_vector_type(2))) float v2f;
typedef __attribute__((ext_vector_type(8))) float v8f;

namespace {
constexpr int kNP = 150000;
constexpr int kEP = 1200000;
constexpr int kNL = 16000;
constexpr int kEL = 64000;
constexpr int kEI = 400000;
constexpr int kNJ = kNP + kNL;
constexpr int kB  = 512;
constexpr int kL  = 3;
constexpr int kND = 58;
constexpr int kED = 6;
constexpr int RELU_FLAG = 1;
constexpr int RES_FLAG  = 2;
constexpr float kNEG_BIG = -3.402823466e38f;
}

// ---------------------------------------------------------------------------
// fill
__global__ void k_fill(float* __restrict__ p, long long n, float v) {
  long long i = (long long)blockIdx.x * blockDim.x + threadIdx.x;
  long long st = (long long)gridDim.x * blockDim.x;
  for (; i < n; i += st) p[i] = v;
}

// ---------------------------------------------------------------------------
// out[M,64] = X[M,K] @ W[K,64]  (small K: 58 or 6), W staged in LDS
__global__ __launch_bounds__(256) void k_embed(
    const float* __restrict__ X, const float* __restrict__ W,
    float* __restrict__ out, int M, int K) {
  __shared__ float Wl[58 * 64];
  for (int i = threadIdx.x; i < K * 64; i += 256) Wl[i] = W[i];
  __syncthreads();
  const int d = threadIdx.x & 63;
  const int local = threadIdx.x >> 6;   // 4 rows per block
  for (int n = blockIdx.x * 4 + local; n < M; n += gridDim.x * 4) {
    const float* x = X + (long long)n * K;
    float acc = 0.f;
    for (int k = 0; k < K; ++k) acc = fmaf(x[k], Wl[k * 64 + d], acc);
    out[(long long)n * 64 + d] = acc;
  }
}

// ---------------------------------------------------------------------------
// out[M,64] = epilogue( A[M,64] @ W[64,64] ), fp32 WMMA 16x16x4.
// One wave owns a 16(M) x 64(N) strip: 4 f32 accumulators, 16 K-steps.
__global__ __launch_bounds__(256) void k_gemm64(
    const float* __restrict__ A, const float* __restrict__ W,
    const float* __restrict__ res, float* __restrict__ out,
    int M, int flags) {
  __shared__ float Wt[64 * 66];            // W transposed, padded (bank spread)
  for (int i = threadIdx.x; i < 64 * 64; i += 256) {
    int r = i >> 6, c = i & 63;
    Wt[c * 66 + r] = W[i];
  }
  __syncthreads();
  const int lane  = threadIdx.x & 31;
  const int wave  = threadIdx.x >> 5;
  const int row   = lane & 15;
  const int khalf = (lane >> 4) << 1;      // 0 (lanes 0-15) / 2 (lanes 16-31)
  const int col   = lane & 15;
  const int mtiles = M >> 4;
  for (int tile = blockIdx.x * 8 + wave; tile < mtiles; tile += gridDim.x * 8) {
    v8f c0 = {}, c1 = {}, c2 = {}, c3 = {};
    const float* Arow = A + (long long)(tile * 16 + row) * 64 + khalf;
#pragma unroll
    for (int kk = 0; kk < 64; kk += 4) {
      v2f a; a.x = Arow[kk]; a.y = Arow[kk + 1];
      const float* wp = Wt + kk + khalf;
      v2f b0; b0.x = wp[(col     ) * 66]; b0.y = wp[(col     ) * 66 + 1];
      v2f b1; b1.x = wp[(col + 16) * 66]; b1.y = wp[(col + 16) * 66 + 1];
      v2f b2; b2.x = wp[(col + 32) * 66]; b2.y = wp[(col + 32) * 66 + 1];
      v2f b3; b3.x = wp[(col + 48) * 66]; b3.y = wp[(col + 48) * 66 + 1];
      c0 = __builtin_amdgcn_wmma_f32_16x16x4_f32(false, a, false, b0, (short)0, c0, false, false);
      c1 = __builtin_amdgcn_wmma_f32_16x16x4_f32(false, a, false, b1, (short)0, c1, false, false);
      c2 = __builtin_amdgcn_wmma_f32_16x16x4_f32(false, a, false, b2, (short)0, c2, false, false);
      c3 = __builtin_amdgcn_wmma_f32_16x16x4_f32(false, a, false, b3, (short)0, c3, false, false);
    }
    const int rbase = tile * 16 + ((lane >> 4) << 3);  // rows r / r+8 per half
#pragma unroll
    for (int r = 0; r < 8; ++r) {
      long long o = (long long)(rbase + r) * 64 + col;
      float v[4] = { c0[r], c1[r], c2[r], c3[r] };
      if (flags & RELU_FLAG) {
#pragma unroll
        for (int t = 0; t < 4; ++t) v[t] = fmaxf(v[t], 0.f);
      }
      if (flags & RES_FLAG) {
#pragma unroll
        for (int t = 0; t < 4; ++t) v[t] += res[o + 16 * t];
      }
#pragma unroll
      for (int t = 0; t < 4; ++t) out[o + 16 * t] = v[t];
    }
  }
}

// ---------------------------------------------------------------------------
// out[dst[e]][d] += h[src[e]][d]
__global__ void k_scatter_add(const float* __restrict__ h,
                              const int* __restrict__ src,
                              const int* __restrict__ dst,
                              float* __restrict__ out, int E) {
  long long total = (long long)E * 64;
  long long st = (long long)gridDim.x * blockDim.x;
  for (long long i = (long long)blockIdx.x * blockDim.x + threadIdx.x; i < total; i += st) {
    int e = (int)(i >> 6), d = (int)(i & 63);
    atomicAdd(&out[(long long)dst[e] * 64 + d], h[(long long)src[e] * 64 + d]);
  }
}

// ---------------------------------------------------------------------------
// GAT edge logits, one wave per edge. rank1: ee_e = dist[e]*wd[0..63], dote=wd[64]
__global__ __launch_bounds__(256) void k_gat_logits(
    const float* __restrict__ x, const float* __restrict__ ee,
    const float* __restrict__ dist, const float* __restrict__ wd,
    const float* __restrict__ a_s, const float* __restrict__ a_d,
    const float* __restrict__ a_e,
    const int* __restrict__ src, const int* __restrict__ dst,
    float* __restrict__ logit, int E, int rank1) {
  int warp  = (int)((blockIdx.x * (long long)blockDim.x + threadIdx.x) >> 5);
  int lane  = threadIdx.x & 31;
  int nwarp = (int)(((long long)gridDim.x * blockDim.x) >> 5);
  for (int e = warp; e < E; e += nwarp) {
    long long s = (long long)src[e] * 64, t = (long long)dst[e] * 64;
    float acc = x[s + lane] * a_s[lane] + x[s + lane + 32] * a_s[lane + 32]
              + x[t + lane] * a_d[lane] + x[t + lane + 32] * a_d[lane + 32];
    if (!rank1) {
      long long eb = (long long)e * 64;
      acc += ee[eb + lane] * a_e[lane] + ee[eb + lane + 32] * a_e[lane + 32];
    }
#pragma unroll
    for (int off = 16; off; off >>= 1) acc += __shfl_xor(acc, off, 32);
    if (lane == 0) {
      if (rank1) acc += dist[e] * wd[64];
      logit[e] = acc > 0.f ? acc : 0.2f * acc;   // leaky_relu(0.2)
    }
  }
}

// ---------------------------------------------------------------------------
__device__ inline void atomicMaxFloat(float* addr, float val) {
  if (val >= 0.f) atomicMax((int*)addr, __float_as_int(val));
  else            atomicMin((unsigned int*)addr, __float_as_uint(val));
}

__global__ void k_seg_max(const float* __restrict__ logit,
                          const int* __restrict__ dst,
                          float* __restrict__ m, int E) {
  long long st = (long long)gridDim.x * blockDim.x;
  for (long long e = (long long)blockIdx.x * blockDim.x + threadIdx.x; e < E; e += st)
    atomicMaxFloat(&m[dst[e]], logit[e]);
}

__global__ void k_exp_sum(const float* __restrict__ logit,
                          const float* __restrict__ m,
                          const int* __restrict__ dst,
                          float* __restrict__ ex, float* __restrict__ ssum, int E) {
  long long st = (long long)gridDim.x * blockDim.x;
  for (long long e = (long long)blockIdx.x * blockDim.x + threadIdx.x; e < E; e += st) {
    int t = dst[e];
    float v = expf(logit[e] - m[t]);
    ex[e] = v;
    atomicAdd(&ssum[t], v);
  }
}

// out[dst][d] += alpha * (x[src][d] + ee_e[d])
__global__ void k_gat_agg(const float* __restrict__ ex, const float* __restrict__ ssum,
                          const float* __restrict__ x, const float* __restrict__ ee,
                          const float* __restrict__ dist, const float* __restrict__ wd,
                          const int* __restrict__ src, const int* __restrict__ dst,
                          float* __restrict__ out, int E, int rank1) {
  long long total = (long long)E * 64;
  long long st = (long long)gridDim.x * blockDim.x;
  for (long long i = (long long)blockIdx.x * blockDim.x + threadIdx.x; i < total; i += st) {
    int e = (int)(i >> 6), d = (int)(i & 63);
    int t = dst[e];
    float alpha = ex[e] / (ssum[t] + 1e-9f);
    float eev = rank1 ? dist[e] * wd[d] : ee[(long long)e * 64 + d];
    float val = x[(long long)src[e] * 64 + d] + eev;
    atomicAdd(&out[(long long)t * 64 + d], alpha * val);
  }
}

// h = relu(out) + h
__global__ void k_relu_res(const float* __restrict__ out, float* __restrict__ h,
                           long long n) {
  long long st = (long long)gridDim.x * blockDim.x;
  for (long long i = (long long)blockIdx.x * blockDim.x + threadIdx.x; i < n; i += st)
    h[i] = fmaxf(out[i], 0.f) + h[i];
}

// ---------------------------------------------------------------------------
__global__ void k_dist(const float* __restrict__ coord,
                       const int* __restrict__ src, const int* __restrict__ dst,
                       float* __restrict__ dist, int E) {
  long long st = (long long)gridDim.x * blockDim.x;
  for (long long e = (long long)blockIdx.x * blockDim.x + threadIdx.x; e < E; e += st) {
    const float* a = coord + (long long)src[e] * 3;
    const float* b = coord + (long long)dst[e] * 3;
    float dx = a[0] - b[0], dy = a[1] - b[1], dz = a[2] - b[2];
    dist[e] = sqrtf(dx * dx + dy * dy + dz * dz + 1e-12f);
  }
}

// wd[d] = sum_k Wie[k]*Ae[k][d]; wd[64] = dot(wd[0..63], a_e)   (rank-1 vdw)
__global__ void k_wrow(const float* __restrict__ Wie, const float* __restrict__ Ae,
                       const float* __restrict__ a_e, float* __restrict__ wd) {
  int d = threadIdx.x;
  float acc = 0.f;
  for (int k = 0; k < 64; ++k) acc = fmaf(Wie[k], Ae[k * 64 + d], acc);
  wd[d] = acc;
  __syncthreads();
  if (d == 0) {
    float s = 0.f;
    for (int i = 0; i < 64; ++i) s = fmaf(wd[i], a_e[i], s);
    wd[64] = s;
  }
}

// readout[gid[n]][d] += hj[n][d]
__global__ void k_readout(const float* __restrict__ hj, const int* __restrict__ gid,
                          float* __restrict__ ro, int N) {
  long long total = (long long)N * 64;
  long long st = (long long)gridDim.x * blockDim.x;
  for (long long i = (long long)blockIdx.x * blockDim.x + threadIdx.x; i < total; i += st) {
    int n = (int)(i >> 6), d = (int)(i & 63);
    atomicAdd(&ro[(long long)gid[n] * 64 + d], hj[i]);
  }
}

// out[r] = dot(t1[r,:], W2[:,0]); one wave per row
__global__ __launch_bounds__(256) void k_final(const float* __restrict__ t1,
                                               const float* __restrict__ W2,
                                               float* __restrict__ out, int R) {
  int warp  = (int)((blockIdx.x * (long long)blockDim.x + threadIdx.x) >> 5);
  int lane  = threadIdx.x & 31;
  int nwarp = (int)(((long long)gridDim.x * blockDim.x) >> 5);
  for (int r = warp; r < R; r += nwarp) {
    long long rb = (long long)r * 64;
    float acc = t1[rb + lane] * W2[lane] + t1[rb + lane + 32] * W2[lane + 32];
#pragma unroll
    for (int off = 16; off; off >>= 1) acc += __shfl_xor(acc, off, 32);
    if (lane == 0) out[r] = acc;
  }
}

// ---------------------------------------------------------------------------
static inline unsigned gsz(long long total, int bs, long long cap = 16384) {
  long long b = (total + bs - 1) / bs;
  if (b < 1) b = 1;
  if (b > cap) b = cap;
  return (unsigned)b;
}

extern "C" void kernel_launch(void* const* d_in, const int* in_sizes, int n_in,
                              void* d_out, int out_size, void* d_ws, size_t ws_size,
                              hipStream_t stream) {
  (void)in_sizes; (void)n_in; (void)out_size; (void)ws_size;
  const float* h_p   = (const float*)d_in[0];
  // d_in[1] (e_p) is dead in the reference: ep = e_p @ We_p is never used.
  const int*   src_p = (const int*)d_in[2];
  const int*   dst_p = (const int*)d_in[3];
  const float* h_l   = (const float*)d_in[4];
  const float* e_l   = (const float*)d_in[5];
  const int*   src_l = (const int*)d_in[6];
  const int*   dst_l = (const int*)d_in[7];
  const float* coord = (const float*)d_in[8];
  const int*   src_i = (const int*)d_in[9];
  const int*   dst_i = (const int*)d_in[10];
  const int*   gid_j = (const int*)d_in[11];
  const float* Wn_p  = (const float*)d_in[12];
  const float* Wn_l  = (const float*)d_in[14];
  const float* We_l  = (const float*)d_in[15];
  const float* Wemb_in = (const float*)d_in[16];
  const float* Wemb_ie = (const float*)d_in[17];
  const float* Wc    = (const float*)d_in[18];
  const float* Al_h  = (const float*)d_in[19];
  const float* Al_e  = (const float*)d_in[20];
  const float* al_s  = (const float*)d_in[21];
  const float* al_d  = (const float*)d_in[22];
  const float* al_e  = (const float*)d_in[23];
  const float* Ai_h  = (const float*)d_in[24];
  const float* Ai_e  = (const float*)d_in[25];
  const float* ai_s  = (const float*)d_in[26];
  const float* ai_d  = (const float*)d_in[27];
  const float* ai_e  = (const float*)d_in[28];
  const float* W1    = (const float*)d_in[29];
  const float* W2    = (const float*)d_in[30];
  float* outp = (float*)d_out;

  // workspace layout
  float* ws = (float*)d_ws;
  long long off = 0;
  auto alloc = [&](long long n) { float* p = ws + off; off += n; return p; };
  float* hp    = alloc((long long)kNP * 64);
  float* hl    = alloc((long long)kNL * 64);
  float* hj    = alloc((long long)kNJ * 64);
  float* el    = alloc((long long)kEL * 64);
  float* agg   = alloc((long long)kNJ * 64);   // GCN agg / GAT out buffer
  float* xbuf  = alloc((long long)kNJ * 64);   // GAT x = h @ Wh
  float* eebuf = alloc((long long)kEL * 64);   // ligand ee per layer
  float* logit = alloc(kEI);
  float* exb   = alloc(kEI);
  float* mbuf  = alloc(kNJ);
  float* sbuf  = alloc(kNJ);
  float* dist  = alloc(kEI);
  float* wd    = alloc(80);                    // wrow[64] + dote
  float* ro    = alloc((long long)kB * 64);
  float* t1    = alloc((long long)kB * 64);

  auto gemm = [&](const float* A, const float* W, const float* res, float* o,
                  int M, int flags) {
    unsigned blocks = gsz((long long)(M >> 4), 8, 4096);
    k_gemm64<<<blocks, 256, 0, stream>>>(A, W, res, o, M, flags);
  };

  // ---- embeddings (ep is dead code in the reference; skipped) ----
  k_embed<<<gsz(kNP, 4, 8192), 256, 0, stream>>>(h_p, Wn_p, hp, kNP, kND);
  k_embed<<<gsz(kNL, 4, 8192), 256, 0, stream>>>(h_l, Wn_l, hl, kNL, kND);
  k_embed<<<gsz(kEL, 4, 8192), 256, 0, stream>>>(e_l, We_l, el, kEL, kED);

  // ---- protein branch: 3x GCN ----
  for (int i = 0; i < kL; ++i) {
    k_fill<<<gsz((long long)kNP * 64, 256), 256, 0, stream>>>(agg, (long long)kNP * 64, 0.f);
    k_scatter_add<<<gsz((long long)kEP * 64, 256), 256, 0, stream>>>(hp, src_p, dst_p, agg, kEP);
    gemm(agg, Wc + (long long)i * 64 * 64, hp, hp, kNP, RELU_FLAG | RES_FLAG);
  }

  // ---- ligand branch: 3x GAT (dense edge features) ----
  for (int i = 0; i < kL; ++i) {
    const float* Wh = Al_h + (long long)i * 64 * 64;
    const float* We = Al_e + (long long)i * 64 * 64;
    gemm(hl, Wh, nullptr, xbuf, kNL, 0);
    gemm(el, We, nullptr, eebuf, kEL, 0);
    k_gat_logits<<<gsz((long long)kEL * 32, 256), 256, 0, stream>>>(
        xbuf, eebuf, nullptr, nullptr, al_s + i * 64, al_d + i * 64, al_e + i * 64,
        src_l, dst_l, logit, kEL, 0);
    k_fill<<<gsz(kNL, 256), 256, 0, stream>>>(mbuf, kNL, kNEG_BIG);
    k_seg_max<<<gsz(kEL, 256), 256, 0, stream>>>(logit, dst_l, mbuf, kEL);
    k_fill<<<gsz(kNL, 256), 256, 0, stream>>>(sbuf, kNL, 0.f);
    k_exp_sum<<<gsz(kEL, 256), 256, 0, stream>>>(logit, mbuf, dst_l, exb, sbuf, kEL);
    k_fill<<<gsz((long long)kNL * 64, 256), 256, 0, stream>>>(agg, (long long)kNL * 64, 0.f);
    k_gat_agg<<<gsz((long long)kEL * 64, 256), 256, 0, stream>>>(
        exb, sbuf, xbuf, eebuf, nullptr, nullptr, src_l, dst_l, agg, kEL, 0);
    k_relu_res<<<gsz((long long)kNL * 64, 256), 256, 0, stream>>>(agg, hl, (long long)kNL * 64);
  }

  // ---- join: hj = concat(hp, hl) @ Wemb_in ; pair distances ----
  gemm(hp, Wemb_in, nullptr, hj, kNP, 0);
  gemm(hl, Wemb_in, nullptr, hj + (long long)kNP * 64, kNL, 0);
  k_dist<<<gsz(kEI, 256), 256, 0, stream>>>(coord, src_i, dst_i, dist, kEI);

  // ---- interaction: 3x GAT with rank-1 edge features (vdw = dist @ Wemb_ie) ----
  for (int i = 0; i < kL; ++i) {
    k_wrow<<<1, 64, 0, stream>>>(Wemb_ie, Ai_e + (long long)i * 64 * 64, ai_e + i * 64, wd);
    gemm(hj, Ai_h + (long long)i * 64 * 64, nullptr, xbuf, kNJ, 0);
    k_gat_logits<<<gsz((long long)kEI * 32, 256), 256, 0, stream>>>(
        xbuf, nullptr, dist, wd, ai_s + i * 64, ai_d + i * 64, ai_e + i * 64,
        src_i, dst_i, logit, kEI, 1);
    k_fill<<<gsz(kNJ, 256), 256, 0, stream>>>(mbuf, kNJ, kNEG_BIG);
    k_seg_max<<<gsz(kEI, 256), 256, 0, stream>>>(logit, dst_i, mbuf, kEI);
    k_fill<<<gsz(kNJ, 256), 256, 0, stream>>>(sbuf, kNJ, 0.f);
    k_exp_sum<<<gsz(kEI, 256), 256, 0, stream>>>(logit, mbuf, dst_i, exb, sbuf, kEI);
    k_fill<<<gsz((long long)kNJ * 64, 256), 256, 0, stream>>>(agg, (long long)kNJ * 64, 0.f);
    k_gat_agg<<<gsz((long long)kEI * 64, 256), 256, 0, stream>>>(
        exb, sbuf, xbuf, nullptr, dist, wd, src_i, dst_i, agg, kEI, 1);
    k_relu_res<<<gsz((long long)kNJ * 64, 256), 256, 0, stream>>>(agg, hj, (long long)kNJ * 64);
  }

  // ---- readout + MLP ----
  k_fill<<<gsz((long long)kB * 64, 256), 256, 0, stream>>>(ro, (long long)kB * 64, 0.f);
  k_readout<<<gsz((long long)kNJ * 64, 256), 256, 0, stream>>>(hj, gid_j, ro, kNJ);
  gemm(ro, W1, nullptr, t1, kB, RELU_FLAG);
  k_final<<<gsz((long long)kB * 32, 256), 256, 0, stream>>>(t1, W2, outp, kB);
}